// MeanFieldGaussianLinear_53558242181640
// MI455X (gfx1250) — compile-verified
//
#include <hip/hip_runtime.h>

typedef __attribute__((ext_vector_type(2))) float v2f;
typedef __attribute__((ext_vector_type(4))) float v4f;
typedef __attribute__((ext_vector_type(8))) float v8f;

#define IN_F   4096
#define OUT_F  4096
#define BATCH  32
#define KT     64          // K-chunk staged per wave per iteration
#define LDS_STRIDE 68      // 64 + 4 dword pad: conflict-free b64 fragment reads, 272B (16B-aligned) rows
#define WPB    4           // waves per block

// 3-array staging registers for one 16-row x 64-col chunk (8 x float4 per array per lane)
struct Stage3 {
    v4f m[8], s[8], e[8];
};

__device__ __forceinline__ void load_stage(const float* __restrict__ mw,
                                           const float* __restrict__ sw,
                                           const float* __restrict__ ew,
                                           size_t base, Stage3& f) {
    // slot s covers rows {2s + lane/16}, cols (lane%16)*4 .. +3  -> 256B contiguous per 16 lanes
#pragma unroll
    for (int s = 0; s < 8; ++s) {
        size_t off = base + (size_t)s * 2 * IN_F;
        f.m[s] = *(const v4f*)(mw + off);
        f.s[s] = *(const v4f*)(sw + off);
        f.e[s] = *(const v4f*)(ew + off);
    }
}

__device__ __forceinline__ void combine_store(float* __restrict__ zdst, int soff, const Stage3& f) {
#pragma unroll
    for (int s = 0; s < 8; ++s) {
        v4f z;
        z.x = f.m[s].x + __expf(f.s[s].x) * f.e[s].x;
        z.y = f.m[s].y + __expf(f.s[s].y) * f.e[s].y;
        z.z = f.m[s].z + __expf(f.s[s].z) * f.e[s].z;
        z.w = f.m[s].w + __expf(f.s[s].w) * f.e[s].w;
        *(v4f*)(zdst + soff + s * 2 * LDS_STRIDE) = z;
    }
}

// grid.x = OUT_F/16/WPB = 64, grid.y = K-split count; block = 128 threads (4 wave32)
// Each wave owns one 16-column output tile and a private LDS staging region: no barriers needed.
__global__ void __launch_bounds__(128, 1)
mfgl_gemm(const float* __restrict__ x,
          const float* __restrict__ mean_w, const float* __restrict__ sigma_w,
          const float* __restrict__ mean_b, const float* __restrict__ sigma_b,
          const float* __restrict__ eps_w,  const float* __restrict__ eps_b,
          float* __restrict__ outp, int kPer) {
    __shared__ float zs_all[WPB * 16 * LDS_STRIDE];

    const int lane = threadIdx.x & 31;
    const int wave = threadIdx.x >> 5;
    const int hi   = lane >> 4;        // half-wave id (K sub-offset selector)
    const int lo   = lane & 15;        // row / column-in-tile id
    const int n0   = (blockIdx.x * WPB + wave) * 16;
    const int kBegin = blockIdx.y * kPer;
    const int nChunks = kPer / KT;

    float* zwave = zs_all + wave * 16 * LDS_STRIDE;

    // staging addresses: row = 2s + hi, col = lo*4 within chunk
    size_t gbase = (size_t)(n0 + hi) * IN_F + (size_t)lo * 4 + kBegin;
    const int soff = hi * LDS_STRIDE + lo * 4;

    // fragment-read addresses
    const float* xb  = x + (size_t)lo * IN_F + hi * 2;            // A: m=lo (+16), k = kc + kk + hi*2
    const float* zrb = zwave + lo * LDS_STRIDE + hi * 2;          // B: row n=lo, k = kk + hi*2

    v8f acc0 = {0.f, 0.f, 0.f, 0.f, 0.f, 0.f, 0.f, 0.f};
    v8f acc1 = {0.f, 0.f, 0.f, 0.f, 0.f, 0.f, 0.f, 0.f};

    Stage3 cur;
    load_stage(mean_w, sigma_w, eps_w, gbase, cur);

    for (int c = 0; c < nChunks; ++c) {
        // fuse z = mean + exp(sigma)*eps, stash in LDS (wave-private, in-order vs our reads)
        combine_store(zwave, soff, cur);
        // prefetch next chunk's 24 B128 loads before the WMMA burst
        if (c + 1 < nChunks)
            load_stage(mean_w, sigma_w, eps_w, gbase + (size_t)(c + 1) * KT, cur);

        const int kc = kBegin + c * KT;
#pragma unroll
        for (int ks = 0; ks < KT / 4; ++ks) {
            const int kk = ks * 4;
            v2f b  = *(const v2f*)(zrb + kk);                       // W-tile fragment (4x16)
            v2f a0 = *(const v2f*)(xb + kc + kk);                   // x rows 0..15
            v2f a1 = *(const v2f*)(xb + (size_t)16 * IN_F + kc + kk); // x rows 16..31
            acc0 = __builtin_amdgcn_wmma_f32_16x16x4_f32(false, a0, false, b,
                                                         (short)0, acc0, false, false);
            acc1 = __builtin_amdgcn_wmma_f32_16x16x4_f32(false, a1, false, b,
                                                         (short)0, acc1, false, false);
        }
    }

    // bias, folded into K-slab 0 only
    float zb = 0.f;
    if (blockIdx.y == 0) {
        int n = n0 + lo;
        zb = mean_b[n] + __expf(sigma_b[n]) * eps_b[n];
    }

    // D layout: VGPR v, lane L -> m = v + 8*(L>>4), n = L&15
    float* op = outp + (size_t)blockIdx.y * (BATCH * OUT_F) + n0 + lo;
#pragma unroll
    for (int v = 0; v < 8; ++v) {
        int m = v + hi * 8;
        op[(size_t)m * OUT_F]        = acc0[v] + zb;
        op[(size_t)(m + 16) * OUT_F] = acc1[v] + zb;
    }
}

__global__ void mfgl_reduce(const float* __restrict__ p, float* __restrict__ out,
                            int n, int ks) {
    int i = blockIdx.x * 256 + threadIdx.x;
    if (i >= n) return;
    float s = 0.f;
    for (int j = 0; j < ks; ++j) s += p[(size_t)j * n + i];
    out[i] = s;
}

extern "C" void kernel_launch(void* const* d_in, const int* in_sizes, int n_in,
                              void* d_out, int out_size, void* d_ws, size_t ws_size,
                              hipStream_t stream) {
    (void)in_sizes; (void)n_in; (void)out_size;
    const float* x       = (const float*)d_in[0];
    const float* mean_w  = (const float*)d_in[1];
    const float* sigma_w = (const float*)d_in[2];
    const float* mean_b  = (const float*)d_in[3];
    const float* sigma_b = (const float*)d_in[4];
    const float* eps_w   = (const float*)d_in[5];
    const float* eps_b   = (const float*)d_in[6];

    const size_t slab = (size_t)BATCH * OUT_F * sizeof(float);   // 512 KB per K-partial
    int KS = 1;
    if      (ws_size >= 8 * slab) KS = 8;
    else if (ws_size >= 4 * slab) KS = 4;
    else if (ws_size >= 2 * slab) KS = 2;

    float* outp = (KS > 1) ? (float*)d_ws : (float*)d_out;

    dim3 grid(OUT_F / 16 / WPB, KS);   // 64 x KS blocks
    dim3 block(WPB * 32);              // 128 threads = 4 wave32
    mfgl_gemm<<<grid, block, 0, stream>>>(x, mean_w, sigma_w, mean_b, sigma_b,
                                          eps_w, eps_b, outp, IN_F / KS);
    if (KS > 1) {
        const int n = BATCH * OUT_F;
        mfgl_reduce<<<(n + 255) / 256, 256, 0, stream>>>((const float*)d_ws,
                                                         (float*)d_out, n, KS);
    }
}